// GraphInfer_55954833932749
// MI455X (gfx1250) — compile-verified
//
#include <hip/hip_runtime.h>

typedef __attribute__((ext_vector_type(2))) float v2f;
typedef __attribute__((ext_vector_type(8))) float v8f;

#define SLOT_DIM 128
#define HIDDEN   64
#define NB       64
#define NK       128
// LR row length = 2*HIDDEN = 128

// ---------------------------------------------------------------------------
// Kernel A: LR[g, n] = concat( slots @ W1[:, :128]^T , slots @ W1[:, 128:]^T )
//   slots viewed as (8192, 128) row-major, LR is (8192, 128) row-major.
//   One wave32 computes one 16x16 tile with 32x V_WMMA_F32_16X16X4_F32.
// ---------------------------------------------------------------------------
__global__ __launch_bounds__(256) void proj_gemm_wmma(
    const float* __restrict__ slots,   // (8192, 128)
    const float* __restrict__ W1,      // (64, 256) row-major
    float* __restrict__ LR)            // (8192, 128)
{
    const int lane  = threadIdx.x & 31;
    const int wave  = threadIdx.x >> 5;
    const int mtile = blockIdx.x * 8 + wave;   // 0..511  (M = 8192)
    const int ntile = blockIdx.y;              // 0..7    (N = 128)

    // f32 16x4 A layout: lanes 0-15 hold M=lane, K=kbase{0,1};
    //                    lanes 16-31 hold M=lane-16, K=kbase{2,3}.
    const int m15 = lane & 15;
    const int kb  = (lane >> 4) << 1;          // 0 or 2

    const int    gRow = mtile * 16 + m15;
    const float* aptr = slots + (size_t)gRow * SLOT_DIM;

    // B[kk][n] = (n<64) ? W1[n][kk] : W1[n-64][128+kk]
    const int    n    = ntile * 16 + m15;
    const int    wrow = (ntile >= 4) ? (n - HIDDEN) : n;
    const float* bptr = W1 + (size_t)wrow * (2 * SLOT_DIM) + ((ntile >= 4) ? SLOT_DIM : 0);

    v8f c = {};
#pragma unroll
    for (int k0 = 0; k0 < SLOT_DIM; k0 += 4) {
        v2f a, b;
        a.x = aptr[k0 + kb];
        a.y = aptr[k0 + kb + 1];
        b.x = bptr[k0 + kb];
        b.y = bptr[k0 + kb + 1];
        // 8 args: (neg_a, A, neg_b, B, c_mod, C, reuse_a, reuse_b)
        c = __builtin_amdgcn_wmma_f32_16x16x4_f32(false, a, false, b,
                                                  (short)0, c, false, false);
    }

    // C/D 16x16 f32 layout: VGPR r -> row r (lanes 0-15) / row r+8 (lanes 16-31)
#pragma unroll
    for (int r = 0; r < 8; ++r) {
        const int mo = r + ((lane >> 4) << 3);
        LR[(size_t)(mtile * 16 + mo) * 128 + ntile * 16 + m15] = c[r];
    }
}

// ---------------------------------------------------------------------------
// Kernel B: out[b,i,j] = (i==j) ? 0 :
//     sigmoid( sum_h relu(L[b,i,h] + R[b,j,h] + b1[h]) * W2[h] + b2 )
//   Workgroup = (b, 16-row i tile). L tile + all R rows staged in LDS.
// ---------------------------------------------------------------------------
__global__ __launch_bounds__(256) void edge_score(
    const float* __restrict__ LR,      // (8192, 128): [L | R]
    const float* __restrict__ b1,      // (64,)
    const float* __restrict__ W2,      // (64,)
    const float* __restrict__ b2p,     // (1,)
    float* __restrict__ out)           // (64, 128, 128)
{
    __shared__ float Lsh[16 * 64];
    __shared__ float Rsh[128 * 65];    // pad to 65 -> conflict-free column reads
    __shared__ float b1sh[64];
    __shared__ float W2sh[64];

    const int bidx = blockIdx.x;       // batch
    const int i0   = blockIdx.y * 16;  // i tile base
    const int tid  = threadIdx.x;

    const float* base = LR + (size_t)bidx * NK * 128;

    for (int idx = tid; idx < 16 * 64; idx += 256) {
        const int il = idx >> 6, h = idx & 63;
        Lsh[idx] = base[(size_t)(i0 + il) * 128 + h];
    }
    for (int idx = tid; idx < 128 * 64; idx += 256) {
        const int j = idx >> 6, h = idx & 63;
        Rsh[j * 65 + h] = base[(size_t)j * 128 + 64 + h];
    }
    if (tid < 64) {
        b1sh[tid] = b1[tid];
        W2sh[tid] = W2[tid];
    }
    __syncthreads();

    const float b2 = b2p[0];

    for (int e = tid; e < 16 * 128; e += 256) {
        const int il = e >> 7;          // wave-uniform -> Lsh reads broadcast
        const int j  = e & 127;         // consecutive lanes -> coalesced stores
        const float* Lr = Lsh + il * 64;
        const float* Rr = Rsh + j * 65;

        float acc = b2;
#pragma unroll
        for (int h = 0; h < 64; ++h) {
            const float v = Lr[h] + Rr[h] + b1sh[h];
            acc = __builtin_fmaf(fmaxf(v, 0.0f), W2sh[h], acc);
        }
        const float sig = 1.0f / (1.0f + __expf(-acc));
        const int gi = i0 + il;
        out[((size_t)bidx * NK + gi) * NK + j] = (gi == j) ? 0.0f : sig;
    }
}

extern "C" void kernel_launch(void* const* d_in, const int* in_sizes, int n_in,
                              void* d_out, int out_size, void* d_ws, size_t ws_size,
                              hipStream_t stream) {
    (void)in_sizes; (void)n_in; (void)out_size; (void)ws_size;
    const float* slots = (const float*)d_in[0];   // (64,128,128)
    const float* W1    = (const float*)d_in[1];   // (64,256)
    const float* b1    = (const float*)d_in[2];   // (64,)
    const float* W2    = (const float*)d_in[3];   // (64,)
    const float* b2    = (const float*)d_in[4];   // (1,)
    float*       out   = (float*)d_out;           // (64,128,128)
    float*       LR    = (float*)d_ws;            // (8192,128) = 4 MB scratch

    dim3 gA(64, 8);   // 64 blocks x 8 waves = 512 M-tiles; y = 8 N-tiles
    proj_gemm_wmma<<<gA, 256, 0, stream>>>(slots, W1, LR);

    dim3 gB(64, 8);   // 64 batches x 8 i-tiles
    edge_score<<<gB, 256, 0, stream>>>(LR, b1, W2, b2, out);
}